// MGATLayer_26903675142556
// MI455X (gfx1250) — compile-verified
//
#include <hip/hip_runtime.h>

typedef __attribute__((ext_vector_type(2))) float v2f;
typedef __attribute__((ext_vector_type(8))) float v8f;
typedef __attribute__((ext_vector_type(4))) int v4i;

#define Nn 3072
#define Ff 128
#define Hh 8
#define Dd 8
#define Rr 2
#define HD 64           // H*D
#define TJ 64           // j-tile width in the fused kernel
#define NEG_BIG (-1.0e10f)

#if defined(__has_builtin)
#if __has_builtin(__builtin_amdgcn_global_load_async_to_lds_b128)
#define HAVE_ASYNC_LDS 1
#endif
#endif
#ifndef HAVE_ASYNC_LDS
#define HAVE_ASYNC_LDS 0
#endif

// 16-byte global->LDS staging: async (ASYNCcnt-tracked) when available.
__device__ __forceinline__ void stage16(const float* __restrict__ g,
                                        float* __restrict__ l) {
#if HAVE_ASYNC_LDS
  __builtin_amdgcn_global_load_async_to_lds_b128(
      (__attribute__((address_space(1))) v4i*)g,
      (__attribute__((address_space(3))) v4i*)l, 0, 0);
#else
  *(float4*)l = *(const float4*)g;
#endif
}

__device__ __forceinline__ void stage_wait() {
#if HAVE_ASYNC_LDS
#if defined(__has_builtin) && __has_builtin(__builtin_amdgcn_s_wait_asynccnt)
  __builtin_amdgcn_s_wait_asynccnt(0);
#else
  asm volatile("s_wait_asynccnt 0x0" ::: "memory");
#endif
#endif
}

// ---------------------------------------------------------------------------
// Kernel 1: features = X @ W   ([3072,128] @ [128,64])
// One wave per 16x16 output tile, K-loop of V_WMMA_F32_16X16X4_F32.
// ---------------------------------------------------------------------------
__global__ __launch_bounds__(32) void mgat_proj_kernel(
    const float* __restrict__ X, const float* __restrict__ W,
    float* __restrict__ feat) {
  const int lane = threadIdx.x & 31;
  const int lo = lane & 15, half = lane >> 4;
  const int i0 = blockIdx.x * 16, n0 = blockIdx.y * 16;
  const float* xrow = X + (size_t)(i0 + lo) * Ff;
  v8f acc = {};
  for (int kk = 0; kk < Ff; kk += 4) {
    const int k0 = kk + 2 * half;
    v2f a, b;
    a.x = xrow[k0];
    a.y = xrow[k0 + 1];
    b.x = W[(size_t)k0 * HD + n0 + lo];
    b.y = W[(size_t)(k0 + 1) * HD + n0 + lo];
    acc = __builtin_amdgcn_wmma_f32_16x16x4_f32(false, a, false, b, (short)0,
                                                acc, false, false);
  }
  for (int v = 0; v < 8; ++v) {
    const int row = v + 8 * half;
    feat[(size_t)(i0 + row) * HD + n0 + lo] = acc[v];
  }
}

// ---------------------------------------------------------------------------
// Kernel 2: attn_self[i,h] = sum_d feat[i,h,d]*aw_self[h,d]  (same for neigh)
// ---------------------------------------------------------------------------
__global__ __launch_bounds__(256) void mgat_attn_kernel(
    const float* __restrict__ feat, const float* __restrict__ aw_self,
    const float* __restrict__ aw_neigh, float* __restrict__ a_self,
    float* __restrict__ a_neigh) {
  const int t = blockIdx.x * blockDim.x + threadIdx.x;
  if (t >= Nn * Hh) return;
  const int i = t / Hh, h = t % Hh;
  const float* f = feat + (size_t)i * HD + h * Dd;
  float ss = 0.f, sn = 0.f;
#pragma unroll
  for (int d = 0; d < Dd; ++d) {
    ss = fmaf(f[d], aw_self[h * Dd + d], ss);
    sn = fmaf(f[d], aw_neigh[h * Dd + d], sn);
  }
  a_self[t] = ss;
  a_neigh[t] = sn;
}

// ---------------------------------------------------------------------------
// Kernel 3: fused mask + leaky-relu + online softmax + aggregation.
// Block = (16-row i-tile, relation r). 8 waves = 8 heads.
// 64-wide j-tiles: adjacency (16x64), features (64x64), neigh (64x8) staged
// to LDS once per tile via async global->LDS B128; softmax state updated
// once per 64 columns; 16x V_WMMA_F32_16X16X4_F32 per tile per head.
// ---------------------------------------------------------------------------
__global__ __launch_bounds__(256) void mgat_fused_kernel(
    const float* __restrict__ A, const float* __restrict__ feat,
    const float* __restrict__ a_self, const float* __restrict__ a_neigh,
    const float* __restrict__ bias, float* __restrict__ out) {
  __shared__ float sA[16 * TJ];    //  4 KB
  __shared__ float sF[TJ * HD];    // 16 KB
  __shared__ float sNg[TJ * Hh];   //  2 KB

  const int tid = threadIdx.x;
  const int h = tid >> 5;          // wave index == head
  const int lane = tid & 31;
  const int lo = lane & 15, half = lane >> 4;
  const int i0 = blockIdx.x * 16;
  const int r = blockIdx.y;
  const float* __restrict__ Ar = A + (size_t)r * Nn * Nn;

  const float self_m = a_self[(size_t)(i0 + lo) * Hh + h];  // row = lo
  float row_max = -INFINITY;
  float row_sum = 0.f;
  v8f acc = {};

  for (int jt = 0; jt < Nn / TJ; ++jt) {
    const int j0 = jt * TJ;
    // ---- stage tiles into LDS in 16B chunks (all 256 threads) ----
    {
      // sA: 16 rows x 64 cols = 256 chunks (1 per thread)
      const int row = tid >> 4, cg = (tid & 15) * 4;
      stage16(&Ar[(size_t)(i0 + row) * Nn + j0 + cg], &sA[row * TJ + cg]);
      // sF: 64 rows x 64 cols = 1024 chunks (4 per thread)
#pragma unroll
      for (int q = 0; q < 4; ++q) {
        const int ch = tid + q * 256;
        const int fr = ch >> 4, fc = (ch & 15) * 4;
        stage16(&feat[(size_t)(j0 + fr) * HD + fc], &sF[fr * HD + fc]);
      }
      // sNg: 64x8 = 512 floats contiguous = 128 chunks (threads 0..127)
      if (tid < 128) stage16(&a_neigh[(size_t)j0 * Hh + tid * 4], &sNg[tid * 4]);
      if (jt + 1 < Nn / TJ)  // L2 prefetch of next adjacency tile
        __builtin_prefetch(&Ar[(size_t)(i0 + row) * Nn + j0 + TJ + cg], 0, 0);
    }
    stage_wait();
    __syncthreads();

    // ---- scores for this head: s[c][u] covers k = 4c + 2*half + u ----
    float s[TJ / 4][2];
    float tmax = -INFINITY;
#pragma unroll
    for (int c = 0; c < TJ / 4; ++c)
#pragma unroll
      for (int u = 0; u < 2; ++u) {
        const int k = 4 * c + 2 * half + u;
        float t = self_m + sNg[k * Hh + h];
        t = t > 0.f ? t : 0.2f * t;                 // leaky_relu(0.2)
        if (!(sA[lo * TJ + k] > 0.f)) t += NEG_BIG; // adjacency mask
        s[c][u] = t;
        tmax = fmaxf(tmax, t);
      }
    // row max across the lane pair (lane, lane^16) holding the same row
    tmax = fmaxf(tmax, __shfl_xor(tmax, 16, 32));
    const float new_max = fmaxf(row_max, tmax);
    const float corr = __expf(row_max - new_max);
    float psum = 0.f;
#pragma unroll
    for (int c = 0; c < TJ / 4; ++c)
#pragma unroll
      for (int u = 0; u < 2; ++u) {
        s[c][u] = __expf(s[c][u] - new_max);
        psum += s[c][u];
      }
    psum += __shfl_xor(psum, 16, 32);
    row_sum = row_sum * corr + psum;
    row_max = new_max;

    // rescale accumulator once per 64-tile: C vgpr v is row (v + 8*half)
#pragma unroll
    for (int v = 0; v < 8; ++v) {
      const float cm = __shfl(corr, v + 8 * half, 32);
      acc[v] *= cm;
    }

    // acc += P(16x64) @ F(64x16-padded), K split into 16 chunks of 4
#pragma unroll
    for (int c = 0; c < TJ / 4; ++c) {
      v2f aP, bF;
      aP.x = s[c][0];
      aP.y = s[c][1];
      const int k0 = 4 * c + 2 * half;
      bF.x = (lo < Dd) ? sF[k0 * HD + h * Dd + lo] : 0.f;
      bF.y = (lo < Dd) ? sF[(k0 + 1) * HD + h * Dd + lo] : 0.f;
      acc = __builtin_amdgcn_wmma_f32_16x16x4_f32(false, aP, false, bF,
                                                  (short)0, acc, false, false);
    }
    __syncthreads();
  }

  // ---- finalize: normalize, bias, relu, scatter to [N, R, H*D] ----
  const float bval = (lo < Dd) ? bias[h * Dd + lo] : 0.f;
#pragma unroll
  for (int v = 0; v < 8; ++v) {
    const float rs = __shfl(row_sum, v + 8 * half, 32);
    float val = acc[v] / rs + bval;
    val = fmaxf(val, 0.f);
    if (lo < Dd) {
      const int row = i0 + v + 8 * half;
      out[((size_t)row * Rr + r) * HD + h * Dd + lo] = val;
    }
  }
}

// ---------------------------------------------------------------------------
extern "C" void kernel_launch(void* const* d_in, const int* in_sizes, int n_in,
                              void* d_out, int out_size, void* d_ws,
                              size_t ws_size, hipStream_t stream) {
  const float* X        = (const float*)d_in[0];  // [N, F]
  const float* A        = (const float*)d_in[1];  // [R, N, N]
  const float* W        = (const float*)d_in[2];  // [F, H*D]
  const float* aw_self  = (const float*)d_in[3];  // [1, H, D]
  const float* aw_neigh = (const float*)d_in[4];  // [1, H, D]
  const float* bias     = (const float*)d_in[5];  // [1, H, D]
  float* out = (float*)d_out;                     // [N, R, H*D]

  float* feat    = (float*)d_ws;                  // N*HD
  float* a_self  = feat + (size_t)Nn * HD;        // N*H
  float* a_neigh = a_self + (size_t)Nn * Hh;      // N*H

  mgat_proj_kernel<<<dim3(Nn / 16, HD / 16), 32, 0, stream>>>(X, W, feat);
  mgat_attn_kernel<<<(Nn * Hh + 255) / 256, 256, 0, stream>>>(
      feat, aw_self, aw_neigh, a_self, a_neigh);
  mgat_fused_kernel<<<dim3(Nn / 16, Rr), 256, 0, stream>>>(
      A, feat, a_self, a_neigh, bias, out);
}